// MaxBPNN_14516989461225
// MI455X (gfx1250) — compile-verified
//
#include <hip/hip_runtime.h>

// Max-product BP message passing round for MI455X (gfx1250, wave32).
//
// Roofline: ~60 MFLOP vs ~0.85 GB of traffic -> pure HBM-bandwidth bound
// (~36 us at 23.3 TB/s). No matmul exists in this workload (C=2), so the
// matrix pipes are irrelevant; the optimization targets are:
//   * f32 global atomics resolved at L2 (accumulator set = 80 MB, fits the
//     192 MB global L2),
//   * TH=NT cache policy on all stream-once operands so they do not evict
//     the L2-resident accumulators,
//   * 64/128-bit per-lane loads for full wave32 coalescing.

typedef float v2f __attribute__((ext_vector_type(2)));
typedef float v4f __attribute__((ext_vector_type(4)));
typedef int   v2i __attribute__((ext_vector_type(2)));
typedef int   v4i __attribute__((ext_vector_type(4)));

#define LN_ZERO (-100000000000.0f)

// ---------------------------------------------------------------- zero d_out
__global__ void k_zero4(v4f* __restrict__ p, int n4) {
  int i = blockIdx.x * blockDim.x + threadIdx.x;
  if (i < n4) {
    v4f z = 0.0f;
    p[i] = z;
  }
}

// ----------------------------------------- phase 1: var belief scatter-add
// vb_acc[var_idx[e]][c] += msg[e][c]   (atomic f32, accumulators stay in L2)
__global__ void k_var_scatter(const v2f* __restrict__ msgs,
                              const int* __restrict__ var_idx,
                              float* __restrict__ vb_acc, int E) {
  int e = blockIdx.x * blockDim.x + threadIdx.x;
  if (e >= E) return;
  v2f m = __builtin_nontemporal_load(msgs + e);     // stream-once -> NT
  int v = __builtin_nontemporal_load(var_idx + e);  // stream-once -> NT
  float* dst = vb_acc + 2 * v;
  atomicAdd(dst + 0, m.x);   // global_atomic_add_f32 (no-return)
  atomicAdd(dst + 1, m.y);
}

// ----------------------------------- phase 2: mask + clamp var beliefs
// vb = max(mask!=0 ? LN_ZERO : vb, LN_ZERO); result is re-gathered in phase 3,
// so it is stored with default (RT) policy to stay L2-hot.
__global__ void k_var_finalize(v2f* __restrict__ vb,
                               const v2i* __restrict__ mask, int V) {
  int v = blockIdx.x * blockDim.x + threadIdx.x;
  if (v >= V) return;
  v2f b  = vb[v];
  v2i mk = __builtin_nontemporal_load(mask + v);
  v2f r;
  r.x = (mk.x != 0) ? LN_ZERO : fmaxf(b.x, LN_ZERO);
  r.y = (mk.y != 0) ? LN_ZERO : fmaxf(b.y, LN_ZERO);
  vb[v] = r;
}

// --------------------- phase 3: v2f messages, expand-permute, factor scatter
// v2f[c] = clamp(mask, vb[var][c] - msg[e][c])
// expanded[e][k] = v2f[(scatter[e*4+k] - 4e) >> 1]   (pattern is intra-edge)
// fb_acc[fac][k] += expanded[e][k]                   (atomic f32)
__global__ void k_factor_scatter(const v2f* __restrict__ msgs,
                                 const int* __restrict__ fac_idx,
                                 const int* __restrict__ var_idx,
                                 const v2i* __restrict__ v2fmask,
                                 const v4i* __restrict__ scat,
                                 const v2f* __restrict__ vb,  // L2-hot gather
                                 float* __restrict__ fb_acc, int E) {
  int e = blockIdx.x * blockDim.x + threadIdx.x;
  if (e >= E) return;
  int f  = __builtin_nontemporal_load(fac_idx + e);
  int v  = __builtin_nontemporal_load(var_idx + e);
  v2f m  = __builtin_nontemporal_load(msgs + e);
  v2i mk = __builtin_nontemporal_load(v2fmask + e);
  v4i s  = __builtin_nontemporal_load(scat + e);
  v2f b  = vb[v];  // random gather; RT policy so the 16 MB table hits L2

  float t0 = (mk.x != 0) ? LN_ZERO : fmaxf(b.x - m.x, LN_ZERO);
  float t1 = (mk.y != 0) ? LN_ZERO : fmaxf(b.y - m.y, LN_ZERO);

  int base = e << 2;
  float* dst = fb_acc + ((long long)f << 2);
  atomicAdd(dst + 0, ((s.x - base) >> 1) ? t1 : t0);
  atomicAdd(dst + 1, ((s.y - base) >> 1) ? t1 : t0);
  atomicAdd(dst + 2, ((s.z - base) >> 1) ? t1 : t0);
  atomicAdd(dst + 3, ((s.w - base) >> 1) ? t1 : t0);
}

// ---------------------- phase 4: add potentials, mask + clamp factor beliefs
__global__ void k_factor_finalize(v4f* __restrict__ fb,
                                  const v4f* __restrict__ pot,
                                  const v4i* __restrict__ mask, int F) {
  int f = blockIdx.x * blockDim.x + threadIdx.x;
  if (f >= F) return;
  v4f a  = fb[f];
  v4f p  = __builtin_nontemporal_load(pot + f);
  v4i mk = __builtin_nontemporal_load(mask + f);
  v4f r;
  r.x = (mk.x == 1) ? LN_ZERO : fmaxf(a.x + p.x, LN_ZERO);
  r.y = (mk.y == 1) ? LN_ZERO : fmaxf(a.y + p.y, LN_ZERO);
  r.z = (mk.z == 1) ? LN_ZERO : fmaxf(a.z + p.z, LN_ZERO);
  r.w = (mk.w == 1) ? LN_ZERO : fmaxf(a.w + p.w, LN_ZERO);
  __builtin_nontemporal_store(r, fb + f);  // final output, not re-read
}

extern "C" void kernel_launch(void* const* d_in, const int* in_sizes, int n_in,
                              void* d_out, int out_size, void* d_ws, size_t ws_size,
                              hipStream_t stream) {
  const float* msgs    = (const float*)d_in[0];  // [E,2]
  const float* pot     = (const float*)d_in[1];  // [F,2,2]
  const int*   edgeidx = (const int*)  d_in[2];  // [2,E] (row0=fac, row1=var)
  const int*   fmask   = (const int*)  d_in[3];  // [F,2,2]
  const int*   vfmask  = (const int*)  d_in[4];  // [E,2]
  const int*   vbmask  = (const int*)  d_in[5];  // [V,2]
  const int*   scat    = (const int*)  d_in[6];  // [E*4]
  // d_in[7], d_in[8] are num_vars / num_factors as 1-elem device arrays;
  // derive the sizes from in_sizes instead (no host readback during capture).
  const int E  = in_sizes[0] / 2;
  const int Vn = in_sizes[5] / 2;
  const int Fn = in_sizes[1] / 4;

  float* vb = (float*)d_out;             // var_beliefs   [V,2]
  float* fb = vb + (size_t)Vn * 2;       // factor_beliefs[F,2,2]

  const int* fac_idx = edgeidx;
  const int* var_idx = edgeidx + E;

  const int B = 256;  // 8 wave32 waves per block

  int n4 = out_size / 4;  // out_size = V*2 + F*4, both multiples of 4
  k_zero4<<<(n4 + B - 1) / B, B, 0, stream>>>((v4f*)d_out, n4);

  k_var_scatter<<<(E + B - 1) / B, B, 0, stream>>>(
      (const v2f*)msgs, var_idx, vb, E);

  k_var_finalize<<<(Vn + B - 1) / B, B, 0, stream>>>(
      (v2f*)vb, (const v2i*)vbmask, Vn);

  k_factor_scatter<<<(E + B - 1) / B, B, 0, stream>>>(
      (const v2f*)msgs, fac_idx, var_idx, (const v2i*)vfmask,
      (const v4i*)scat, (const v2f*)vb, fb, E);

  k_factor_finalize<<<(Fn + B - 1) / B, B, 0, stream>>>(
      (v4f*)fb, (const v4f*)pot, (const v4i*)fmask, Fn);
}